// MotivationAttLayer2_46943992545843
// MI455X (gfx1250) — compile-verified
//
#include <hip/hip_runtime.h>
#include <math.h>

#define IN_DIM   128
#define OUT_DIM  64
#define NEG_SLOPE 0.01f

typedef __attribute__((ext_vector_type(2))) float v2f;
typedef __attribute__((ext_vector_type(4))) float v4f;
typedef __attribute__((ext_vector_type(8))) float v8f;

// ---------------------------------------------------------------------------
// Kernel A: fold attention right-half into W_dst:  w_r[k] = sum_o W_dst[o][k] * a_r[o]
// (valid because softplus is never applied to z_dst)
// ---------------------------------------------------------------------------
__global__ void k_wr(const float* __restrict__ Wd, const float* __restrict__ attn,
                     float* __restrict__ wr) {
    int k = threadIdx.x;  // 128 threads
    float s = 0.f;
#pragma unroll 8
    for (int o = 0; o < OUT_DIM; ++o)
        s = fmaf(Wd[o * IN_DIM + k], attn[OUT_DIM + o], s);
    wr[k] = s;
}

// ---------------------------------------------------------------------------
// Kernel B: z_src = h_src @ W_src^T via V_WMMA_F32_16X16X4_F32.
// One wave32 per 16x16 output tile; 32 WMMA ops (K=128 in steps of 4).
// A(16x4) layout: lane L holds float2 A[m][2*(L/16)+{0,1}], m = L%16.
// B(4x16)=W^T   : lane L holds float2 W[n][kk+2*(L/16)+{0,1}], n = L%16.
// C/D           : c[j] -> row m0 + j + 8*(L/16), col n0 + L%16.
// Optional softplus per item_user flag (uniform scalar read).
// ---------------------------------------------------------------------------
__global__ __launch_bounds__(256)
void k_zsrc(const float* __restrict__ h, const float* __restrict__ W,
            const int* __restrict__ item_user, float* __restrict__ z, int nrows) {
    int wave = blockIdx.x * 8 + (threadIdx.x >> 5);
    int lane = threadIdx.x & 31;
    int mt   = wave >> 2;        // M tile (16 rows)
    int nt   = wave & 3;         // N tile (16 of 64 cols)
    int half = lane >> 4;
    int l16  = lane & 15;

    int row  = mt * 16 + l16;
    int rowc = row < nrows ? row : (nrows - 1);   // clamp (keeps EXEC all-1s for WMMA)

    const float* arow = h + (size_t)rowc * IN_DIM + 2 * half;
    const float* brow = W + (size_t)(nt * 16 + l16) * IN_DIM + 2 * half;

    v8f c = {};
#pragma unroll 8
    for (int kk = 0; kk < IN_DIM; kk += 4) {
        v2f a = *(const v2f*)(arow + kk);
        v2f b = *(const v2f*)(brow + kk);
        c = __builtin_amdgcn_wmma_f32_16x16x4_f32(
                /*neg_a=*/false, a, /*neg_b=*/false, b,
                /*c_mod=*/(short)0, c, /*reuse_a=*/false, /*reuse_b=*/false);
    }

    bool sp = (*item_user != 0);
    float* zt = z + (size_t)(mt * 16) * OUT_DIM + nt * 16 + l16;
#pragma unroll
    for (int j = 0; j < 8; ++j) {
        int r = mt * 16 + j + 8 * half;
        if (r < nrows) {
            float v = c[j];
            if (sp) v = fmaxf(v, 0.f) + log1pf(expf(-fabsf(v)));  // stable softplus
            zt[(size_t)(j + 8 * half) * OUT_DIM] = v;
        }
    }
}

// ---------------------------------------------------------------------------
// Kernel C: el[i] = dot(z_src[i,:], attn[0:64]).  One wave32 per row.
// ---------------------------------------------------------------------------
__global__ __launch_bounds__(256)
void k_el(const float* __restrict__ z, const float* __restrict__ attn,
          float* __restrict__ el, int nrows) {
    int wave = blockIdx.x * 8 + (threadIdx.x >> 5);
    int lane = threadIdx.x & 31;
    if (wave >= nrows) return;
    const float* zr = z + (size_t)wave * OUT_DIM;
    float acc = zr[lane] * attn[lane] + zr[lane + 32] * attn[lane + 32];
#pragma unroll
    for (int off = 16; off > 0; off >>= 1) acc += __shfl_xor(acc, off, 32);
    if (lane == 0) el[wave] = acc;
}

// ---------------------------------------------------------------------------
// Kernel D: er[i] = dot(h_dst[i,:128], w_r).  One wave32 per row, float4 loads.
// ---------------------------------------------------------------------------
__global__ __launch_bounds__(256)
void k_er(const float* __restrict__ hd, const float* __restrict__ wr,
          float* __restrict__ er, int nrows) {
    int wave = blockIdx.x * 8 + (threadIdx.x >> 5);
    int lane = threadIdx.x & 31;
    if (wave >= nrows) return;
    v4f hv = *(const v4f*)(hd + (size_t)wave * IN_DIM + lane * 4);
    v4f wv = *(const v4f*)(wr + lane * 4);
    float acc = hv[0] * wv[0] + hv[1] * wv[1] + hv[2] * wv[2] + hv[3] * wv[3];
#pragma unroll
    for (int off = 16; off > 0; off >>= 1) acc += __shfl_xor(acc, off, 32);
    if (lane == 0) er[wave] = acc;
}

// ---------------------------------------------------------------------------
// Init: h_out = 0, m = -inf, denom = 0 (ws/out are poisoned by harness)
// ---------------------------------------------------------------------------
__global__ void k_init(float* __restrict__ out, float* __restrict__ m,
                       float* __restrict__ den, int nout, int ndst) {
    int i = blockIdx.x * blockDim.x + threadIdx.x;
    if (i < nout) out[i] = 0.f;
    if (i < ndst) { m[i] = -__builtin_inff(); den[i] = 0.f; }
}

// Sign-aware float atomic max via integer atomics (works incl. -inf init).
__device__ inline void atomicMaxF(float* addr, float v) {
    if (v >= 0.f) atomicMax((int*)addr, __float_as_int(v));
    else          atomicMin((unsigned int*)addr, __float_as_uint(v));
}

// ---------------------------------------------------------------------------
// Edge pass 1: e = leaky_relu(el[src]+er[dst]); segment max into m[dst]
// ---------------------------------------------------------------------------
__global__ void k_edge_max(const int* __restrict__ src, const int* __restrict__ dst,
                           const float* __restrict__ el, const float* __restrict__ er,
                           float* __restrict__ ebuf, float* __restrict__ m, int ne) {
    int e = blockIdx.x * blockDim.x + threadIdx.x;
    if (e >= ne) return;
    float v = el[src[e]] + er[dst[e]];
    v = v > 0.f ? v : NEG_SLOPE * v;
    ebuf[e] = v;
    atomicMaxF(&m[dst[e]], v);
}

// ---------------------------------------------------------------------------
// Edge pass 2: ex = exp(e - m[dst]); denom[dst] += ex
// ---------------------------------------------------------------------------
__global__ void k_edge_exp(const int* __restrict__ dst, const float* __restrict__ m,
                           float* __restrict__ ebuf, float* __restrict__ den, int ne) {
    int e = blockIdx.x * blockDim.x + threadIdx.x;
    if (e >= ne) return;
    int d = dst[e];
    float ex = expf(ebuf[e] - m[d]);
    ebuf[e] = ex;
    atomicAdd(&den[d], ex);
}

// ---------------------------------------------------------------------------
// Edge pass 3: h_out[dst] += (ex/denom[dst]) * z_src[src].
// One wave32 per edge; lane covers cols {lane, lane+32}. z_src & h_out are
// L2-resident (25.6 MB each in 192 MB L2).
// ---------------------------------------------------------------------------
__global__ __launch_bounds__(256)
void k_scatter(const int* __restrict__ src, const int* __restrict__ dst,
               const float* __restrict__ ebuf, const float* __restrict__ den,
               const float* __restrict__ z, float* __restrict__ out, int ne) {
    int wave = blockIdx.x * 8 + (threadIdx.x >> 5);
    int lane = threadIdx.x & 31;
    if (wave >= ne) return;
    int s = src[wave], d = dst[wave];
    float alpha = ebuf[wave] / den[d];
    const float* zr = z + (size_t)s * OUT_DIM;
    float* orow = out + (size_t)d * OUT_DIM;
    atomicAdd(&orow[lane],      alpha * zr[lane]);
    atomicAdd(&orow[lane + 32], alpha * zr[lane + 32]);
}

// ---------------------------------------------------------------------------
extern "C" void kernel_launch(void* const* d_in, const int* in_sizes, int n_in,
                              void* d_out, int out_size, void* d_ws, size_t ws_size,
                              hipStream_t stream) {
    const float* h_src   = (const float*)d_in[0];
    const float* h_dst   = (const float*)d_in[1];
    const int*   src_idx = (const int*)d_in[2];
    const int*   dst_idx = (const int*)d_in[3];
    const float* W_src   = (const float*)d_in[4];
    const float* W_dst   = (const float*)d_in[5];
    const float* attn    = (const float*)d_in[6];
    const int*   item_u  = (const int*)d_in[7];
    float* out = (float*)d_out;

    int nsrc = in_sizes[0] / IN_DIM;
    int ndst = in_sizes[1] / IN_DIM;
    int ne   = in_sizes[2];

    // workspace carve-out
    char* p = (char*)d_ws;
    float* z    = (float*)p; p += (size_t)nsrc * OUT_DIM * sizeof(float);
    float* el   = (float*)p; p += (size_t)nsrc * sizeof(float);
    float* er   = (float*)p; p += (size_t)ndst * sizeof(float);
    float* m    = (float*)p; p += (size_t)ndst * sizeof(float);
    float* den  = (float*)p; p += (size_t)ndst * sizeof(float);
    float* ebuf = (float*)p; p += (size_t)ne * sizeof(float);
    float* wr   = (float*)p;

    k_wr<<<1, 128, 0, stream>>>(W_dst, attn, wr);

    int mtiles = (nsrc + 15) / 16;
    int waves  = mtiles * 4;                       // 4 N-tiles of 16
    k_zsrc<<<(waves + 7) / 8, 256, 0, stream>>>(h_src, W_src, item_u, z, nsrc);

    k_el<<<(nsrc + 7) / 8, 256, 0, stream>>>(z, attn, el, nsrc);
    k_er<<<(ndst + 7) / 8, 256, 0, stream>>>(h_dst, wr, er, ndst);

    int initN = out_size > ndst ? out_size : ndst;
    k_init<<<(initN + 255) / 256, 256, 0, stream>>>(out, m, den, out_size, ndst);

    k_edge_max<<<(ne + 255) / 256, 256, 0, stream>>>(src_idx, dst_idx, el, er, ebuf, m, ne);
    k_edge_exp<<<(ne + 255) / 256, 256, 0, stream>>>(dst_idx, m, ebuf, den, ne);
    k_scatter<<<(ne + 7) / 8, 256, 0, stream>>>(src_idx, dst_idx, ebuf, den, z, out, ne);
}